// PerTaskEmpiricalNormalization_34497177321524
// MI455X (gfx1250) — compile-verified
//
#include <hip/hip_runtime.h>
#include <stdint.h>

#define NUM_TASKS 64
#define FEAT      128
#define NUM_ENVS  131072
#define EPS       0.01f

#define BLOCKS          256
#define THREADS         256
#define ROWS_PER_BLOCK  (NUM_ENVS / BLOCKS)        // 512
#define TILE_ROWS       64
#define NTILES          (ROWS_PER_BLOCK / TILE_ROWS) // 8
#define TILE_FLOATS     (TILE_ROWS * FEAT)         // 8192 floats = 32KB
#define CHUNKS_PER_THREAD ((TILE_FLOATS * 4) / (THREADS * 16)) // 8 x b128 per thread/tile

// ---- gfx1250 async global->LDS (ASYNCcnt-tracked) -------------------------
__device__ __forceinline__ unsigned lds_offset_u32(const void* p) {
    // LDS aperture occupies addr[63:32]; low 32 bits are the LDS byte offset.
    return (unsigned)(uintptr_t)p;
}

__device__ __forceinline__ void async_load_b128(unsigned lds_byte_off, const float* gptr) {
    asm volatile("global_load_async_to_lds_b128 %0, %1, off"
                 :
                 : "v"(lds_byte_off), "v"((unsigned long long)(uintptr_t)gptr)
                 : "memory");
}

__device__ __forceinline__ void wait_async_le8() {
    asm volatile("s_wait_asynccnt 8" ::: "memory");
}
__device__ __forceinline__ void wait_async_0() {
    asm volatile("s_wait_asynccnt 0" ::: "memory");
}

// ---- zero the global accumulators (ws is poisoned, not zeroed) ------------
__global__ void zero_ws_kernel(float* ws, int n) {
    int i = blockIdx.x * blockDim.x + threadIdx.x;
    if (i < n) ws[i] = 0.0f;
}

// ---- fused: normalize with OLD stats + segmented sum / sumsq / count ------
__global__ __launch_bounds__(THREADS)
void norm_reduce_kernel(const float* __restrict__ x,
                        const int*   __restrict__ task_ids,
                        const float* __restrict__ mean_buf,
                        const float* __restrict__ std_buf,
                        float* __restrict__ out,
                        float* __restrict__ ws_sum,
                        float* __restrict__ ws_sqs,
                        float* __restrict__ ws_cnt) {
    __shared__ float s_sum[NUM_TASKS * FEAT];   // 32 KB
    __shared__ float s_sqs[NUM_TASKS * FEAT];   // 32 KB
    __shared__ float s_cnt[NUM_TASKS];
    __shared__ float s_tile[2][TILE_FLOATS];    // 2 x 32 KB streaming buffers

    const int tid  = threadIdx.x;
    const int bid  = blockIdx.x;
    const int row0 = bid * ROWS_PER_BLOCK;

    for (int i = tid; i < NUM_TASKS * FEAT; i += THREADS) {
        s_sum[i] = 0.0f;
        s_sqs[i] = 0.0f;
    }
    if (tid < NUM_TASKS) s_cnt[tid] = 0.0f;

    // Issue one 32KB tile as 8 async b128 transfers per thread.
    auto issue_tile = [&](int tile, int buf) {
        const float* src = x + (size_t)(row0 + tile * TILE_ROWS) * FEAT;
        const unsigned dst0 = lds_offset_u32(&s_tile[buf][0]);
#pragma unroll
        for (int c = 0; c < CHUNKS_PER_THREAD; ++c) {
            const int chunk16 = c * THREADS + tid;          // 16-byte chunk index
            async_load_b128(dst0 + (unsigned)chunk16 * 16u, src + chunk16 * 4);
        }
    };

    issue_tile(0, 0);

    const int wv   = tid >> 5;   // row slot within tile (8 waves)
    const int lane = tid & 31;   // feature quad: feats [lane*4, lane*4+3]

    for (int t = 0; t < NTILES; ++t) {
        if (t + 1 < NTILES) {
            issue_tile(t + 1, (t + 1) & 1);  // prefetch next tile into other buffer
            wait_async_le8();                // current tile's 8 transfers retired
        } else {
            wait_async_0();
        }
        __syncthreads();                     // data from all waves' async loads visible

        const float* tile = &s_tile[t & 1][0];
#pragma unroll
        for (int k = 0; k < TILE_ROWS / 8; ++k) {
            const int r    = wv + k * 8;
            const int grow = row0 + t * TILE_ROWS + r;
            const int task = task_ids[grow];

            const float4 xv = *(const float4*)&tile[r * FEAT + lane * 4];
            const float4 mv = *(const float4*)&mean_buf[task * FEAT + lane * 4];
            const float4 sv = *(const float4*)&std_buf [task * FEAT + lane * 4];

            float4 ov;
            ov.x = (xv.x - mv.x) / (sv.x + EPS);
            ov.y = (xv.y - mv.y) / (sv.y + EPS);
            ov.z = (xv.z - mv.z) / (sv.z + EPS);
            ov.w = (xv.w - mv.w) / (sv.w + EPS);
            *(float4*)&out[(size_t)grow * FEAT + lane * 4] = ov;

            float* ss = &s_sum[task * FEAT + lane * 4];
            float* sq = &s_sqs[task * FEAT + lane * 4];
            atomicAdd(&ss[0], xv.x); atomicAdd(&ss[1], xv.y);
            atomicAdd(&ss[2], xv.z); atomicAdd(&ss[3], xv.w);
            atomicAdd(&sq[0], xv.x * xv.x); atomicAdd(&sq[1], xv.y * xv.y);
            atomicAdd(&sq[2], xv.z * xv.z); atomicAdd(&sq[3], xv.w * xv.w);
            if (lane == 0) atomicAdd(&s_cnt[task], 1.0f);
        }
        __syncthreads();   // tile buffer free for the prefetch issued next iter
    }

    __syncthreads();
    // One flush per block into L2-resident accumulators (64KB working set).
    for (int i = tid; i < NUM_TASKS * FEAT; i += THREADS) {
        atomicAdd(&ws_sum[i], s_sum[i]);
        atomicAdd(&ws_sqs[i], s_sqs[i]);
    }
    if (tid < NUM_TASKS) atomicAdd(&ws_cnt[tid], s_cnt[tid]);
}

// ---- Chan/Welford merge of running stats (tiny: 64x128) -------------------
__global__ void merge_kernel(const float* __restrict__ ws_sum,
                             const float* __restrict__ ws_sqs,
                             const float* __restrict__ ws_cnt,
                             const float* __restrict__ mean_buf,
                             const float* __restrict__ var_buf,
                             const float* __restrict__ std_buf,
                             const int*   __restrict__ count_buf,
                             float* __restrict__ new_mean,
                             float* __restrict__ new_var,
                             float* __restrict__ new_std,
                             float* __restrict__ new_count) {
    const int i = blockIdx.x * blockDim.x + threadIdx.x;   // 0..8191
    if (i >= NUM_TASKS * FEAT) return;
    const int t = i / FEAT;

    const float n_b     = ws_cnt[t];
    const float safe_nb = fmaxf(n_b, 1.0f);
    const float bm      = ws_sum[i] / safe_nb;
    const float bv      = fmaxf(ws_sqs[i] / safe_nb - bm * bm, 0.0f);

    const float old_n    = (float)count_buf[t];
    const float new_n    = old_n + n_b;
    const float safe_new = fmaxf(new_n, 1.0f);

    const float mean_o = mean_buf[i];
    const float var_o  = var_buf[i];
    const float std_o  = std_buf[i];

    const float delta      = bm - mean_o;
    const float upd_mean   = mean_o + (n_b / safe_new) * delta;
    const float merged_var = (var_o * old_n + bv * n_b +
                              delta * delta * (old_n * n_b / safe_new)) / safe_new;

    const bool has_batch = n_b > 0.0f;
    const bool had_count = old_n > 0.0f;

    const float nv = has_batch ? (had_count ? merged_var : bv) : var_o;
    const float nm = has_batch ? upd_mean : mean_o;
    const float ns = has_batch ? sqrtf(nv) : std_o;

    new_mean[i] = nm;
    new_var[i]  = nv;
    new_std[i]  = ns;
    if ((i & (FEAT - 1)) == 0) new_count[t] = new_n;
}

extern "C" void kernel_launch(void* const* d_in, const int* in_sizes, int n_in,
                              void* d_out, int out_size, void* d_ws, size_t ws_size,
                              hipStream_t stream) {
    const float* x         = (const float*)d_in[0];
    const int*   task_ids  = (const int*)  d_in[1];
    const float* mean_buf  = (const float*)d_in[2];
    const float* var_buf   = (const float*)d_in[3];
    const float* std_buf   = (const float*)d_in[4];
    const int*   count_buf = (const int*)  d_in[5];

    float* out       = (float*)d_out;
    float* new_mean  = out + (size_t)NUM_ENVS * FEAT;
    float* new_var   = new_mean + NUM_TASKS * FEAT;
    float* new_std   = new_var  + NUM_TASKS * FEAT;
    float* new_count = new_std  + NUM_TASKS * FEAT;

    float* ws_sum = (float*)d_ws;
    float* ws_sqs = ws_sum + NUM_TASKS * FEAT;
    float* ws_cnt = ws_sqs + NUM_TASKS * FEAT;

    const int nzero = NUM_TASKS * FEAT * 2 + NUM_TASKS;
    zero_ws_kernel<<<(nzero + 255) / 256, 256, 0, stream>>>((float*)d_ws, nzero);

    norm_reduce_kernel<<<BLOCKS, THREADS, 0, stream>>>(
        x, task_ids, mean_buf, std_buf, out, ws_sum, ws_sqs, ws_cnt);

    merge_kernel<<<(NUM_TASKS * FEAT + 255) / 256, 256, 0, stream>>>(
        ws_sum, ws_sqs, ws_cnt, mean_buf, var_buf, std_buf, count_buf,
        new_mean, new_var, new_std, new_count);
}